// AntiDesigner_61942018343628
// MI455X (gfx1250) — compile-verified
//
#include <hip/hip_runtime.h>
#include <math.h>

#define TOPK 9
#define IDXMASK 0x7FFu   // 11 bits: column index within chain (L <= 2048)

typedef float v2f __attribute__((ext_vector_type(2)));
typedef float v8f __attribute__((ext_vector_type(8)));

struct F3 { float x, y, z; };
__device__ __forceinline__ F3 f3sub(F3 a, F3 b){ return {a.x-b.x, a.y-b.y, a.z-b.z}; }
__device__ __forceinline__ float f3dot(F3 a, F3 b){ return a.x*b.x + a.y*b.y + a.z*b.z; }
__device__ __forceinline__ F3 f3cross(F3 a, F3 b){
  return {a.y*b.z - a.z*b.y, a.z*b.x - a.x*b.z, a.x*b.y - a.y*b.x};
}
__device__ __forceinline__ F3 f3norm(F3 a){
  float n = fmaxf(sqrtf(f3dot(a,a)), 1e-12f);
  return {a.x/n, a.y/n, a.z/n};
}
__device__ __forceinline__ F3 ldAtom(const float* X, int node, int atom){
  const float* p = X + (size_t)node*12 + atom*3;
  return {p[0], p[1], p[2]};
}
// X3[t] = atom (t%3) of node (t/3) -- flattened N/CA/C chain
__device__ __forceinline__ F3 ldX3(const float* X, int t){
  const float* p = X + (size_t)(t/3)*12 + (t%3)*3;
  return {p[0], p[1], p[2]};
}
__device__ __forceinline__ float clip1(float c){
  return fminf(fmaxf(c, -1.0f + 1e-6f), 1.0f - 1e-6f);
}
__device__ __forceinline__ float sgnf(float x){
  return (x > 0.f) ? 1.f : ((x < 0.f) ? -1.f : 0.f);
}
__device__ __forceinline__ void rbf16(float d, float* o){
  #pragma unroll
  for (int j = 0; j < 16; ++j) {
    float t = (d - (20.0f/15.0f)*(float)j) * 0.8f; // sigma = 1.25
    o[j] = expf(-t*t);
  }
}
__device__ __forceinline__ void posembed16(float pos, float* o){
  #pragma unroll
  for (int j = 0; j < 8; ++j) {
    float f = expf(-(float)(2*j) * (9.210340371976184f / 16.0f));
    float a = pos * f;
    o[j]   = cosf(a);
    o[8+j] = sinf(a);
  }
}

// Branchless sorted-ascending top-9 insert on packed (d2|idx) u32 keys.
// new[0]=min(x,v0); new[k]=max(v[k-1], min(x,v[k])) -- pure v_min_u32/v_max_u32,
// dependency depth 2, no EXEC divergence (WMMA needs EXEC all-ones anyway).
__device__ __forceinline__ void insertKey9(unsigned (&v)[TOPK], unsigned x){
  unsigned nv[TOPK];
  nv[0] = min(x, v[0]);
  #pragma unroll
  for (int k = 1; k < TOPK; ++k) nv[k] = max(v[k-1], min(x, v[k]));
  #pragma unroll
  for (int k = 0; k < TOPK; ++k) v[k] = nv[k];
}

// ---------------- Kernel 1: per-node features + frames + CA norms ----------------
__global__ void k_node(const float* __restrict__ X,
                       float* __restrict__ node_pos, float* __restrict__ dis_node,
                       float* __restrict__ angle_node, float* __restrict__ direct_node,
                       float* __restrict__ canorm, float* __restrict__ Ofr,
                       int N, int L)
{
  int i = blockIdx.x * blockDim.x + threadIdx.x;
  if (i >= N) return;

  F3 ca = ldAtom(X, i, 1);
  canorm[i] = f3dot(ca, ca);

  int pl = i % L, chain = i / L, nchain = N / L;

  // node_pos
  {
    float t[16]; posembed16((float)pl, t);
    #pragma unroll
    for (int j = 0; j < 16; ++j) node_pos[(size_t)i*16 + j] = t[j];
  }

  // dis_node: RBF(d_NC), RBF(d_CC), RBF(d_OC)
  {
    const int atoms[3] = {0, 2, 3};
    #pragma unroll
    for (int a = 0; a < 3; ++a) {
      F3 v = f3sub(ldAtom(X, i, atoms[a]), ca);
      float d = sqrtf(f3dot(v, v) + 1e-6f);
      float t[16]; rbf16(d, t);
      #pragma unroll
      for (int j = 0; j < 16; ++j) dis_node[(size_t)i*48 + a*16 + j] = t[j];
    }
  }

  // angle_node: dihedrals + bond angles on the flattened N/CA/C chain
  {
    bool am = !((pl < 2) || (pl == L-1 && chain < nchain-1));
    float amf = am ? 1.f : 0.f;
    int threeN = 3*N;
    float dih[3], ang[3];
    #pragma unroll
    for (int k = 0; k < 3; ++k) {
      int p = 3*i + k;
      float D = 0.f, A = 0.f;
      if (p >= 1 && p <= threeN - 3) {
        F3 x0 = ldX3(X, p-1), x1 = ldX3(X, p), x2 = ldX3(X, p+1), x3 = ldX3(X, p+2);
        F3 u2 = f3norm(f3sub(x1, x0));
        F3 u1 = f3norm(f3sub(x2, x1));
        F3 u0 = f3norm(f3sub(x3, x2));
        F3 n2 = f3norm(f3cross(u2, u1));
        F3 n1 = f3norm(f3cross(u1, u0));
        D = sgnf(f3dot(u2, n1)) * acosf(clip1(f3dot(n2, n1)));
        A = acosf(clip1(f3dot(u2, u1)));
      }
      dih[k] = D; ang[k] = A;
    }
    #pragma unroll
    for (int k = 0; k < 3; ++k) {
      angle_node[(size_t)i*12 + 0 + k] = cosf(dih[k]) * amf;
      angle_node[(size_t)i*12 + 3 + k] = sinf(dih[k]) * amf;
      angle_node[(size_t)i*12 + 6 + k] = cosf(ang[k]) * amf;
      angle_node[(size_t)i*12 + 9 + k] = sinf(ang[k]) * amf;
    }
  }

  // O frame (rows: o1, nf, o1 x nf), zero at pads + frame mask
  float Om[9];
  #pragma unroll
  for (int k = 0; k < 9; ++k) Om[k] = 0.f;
  {
    bool fm = !((pl < 2) || (pl >= L-2 && chain < nchain-1));
    if (fm && i < N-2) {
      F3 cam = ldAtom(X, i-1, 1), cap = ldAtom(X, i+1, 1);
      F3 uc2 = f3norm(f3sub(ca,  cam));
      F3 uc1 = f3norm(f3sub(cap, ca));
      F3 nf  = f3norm(f3cross(uc2, uc1));
      F3 o1  = f3norm(f3sub(uc2, uc1));
      F3 o3  = f3cross(o1, nf);
      Om[0]=o1.x; Om[1]=o1.y; Om[2]=o1.z;
      Om[3]=nf.x; Om[4]=nf.y; Om[5]=nf.z;
      Om[6]=o3.x; Om[7]=o3.y; Om[8]=o3.z;
    }
    #pragma unroll
    for (int k = 0; k < 9; ++k) Ofr[(size_t)i*9 + k] = Om[k];
  }

  // direct_node: normalize(O @ (X[a]-CA)) for a in {0,2,3}
  {
    const int atoms[3] = {0, 2, 3};
    #pragma unroll
    for (int a = 0; a < 3; ++a) {
      F3 v = f3sub(ldAtom(X, i, atoms[a]), ca);
      F3 w = { Om[0]*v.x + Om[1]*v.y + Om[2]*v.z,
               Om[3]*v.x + Om[4]*v.y + Om[5]*v.z,
               Om[6]*v.x + Om[7]*v.y + Om[8]*v.z };
      w = f3norm(w);
      direct_node[(size_t)i*9 + a*3 + 0] = w.x;
      direct_node[(size_t)i*9 + a*3 + 1] = w.y;
      direct_node[(size_t)i*9 + a*3 + 2] = w.z;
    }
  }
}

// ---------------- Kernel 2: fused WMMA distance tiles + running top-9 ----------------
// One wave = 16 query rows of one chain. For each 16-col tile:
//   D2[m][n] = A[m]·B[n] + C[m]  with A[m]=(cx,cy,cz,1), B[n]=(-2qx,-2qy,-2qz,|q|^2), C[m]=|c|^2
// via V_WMMA_F32_16X16X4_F32. D layout: N = lane%16 (query), M = vgpr + 8*(lane>=16) (col).
// Selection runs on d2 directly (sqrt is monotonic; only indices are emitted), using
// packed u32 keys (d2 bits | column index) and a min/max insertion network.
__global__ void __launch_bounds__(256) k_topk(const float* __restrict__ X,
                                              const float* __restrict__ canorm,
                                              float* __restrict__ erow,
                                              float* __restrict__ ecolf,
                                              int N, int L)
{
  int lane  = threadIdx.x & 31;
  int wg    = (blockIdx.x * blockDim.x + threadIdx.x) >> 5;
  int tiles = L >> 4;              // 16-wide column tiles per chain
  int b     = wg / tiles;
  int qt    = wg - b * tiles;
  int half  = lane >> 4;
  int ln    = lane & 15;
  int base  = b * L;
  int qg    = base + qt*16 + ln;   // this lane's query node (N index of tile)

  F3 q = ldAtom(X, qg, 1);
  float qn = f3dot(q, q);
  v2f bf;  // B 4x16: lanes 0-15 hold K=0,1; lanes 16-31 hold K=2,3
  bf.x = half ? (-2.f*q.z) : (-2.f*q.x);
  bf.y = half ? qn         : (-2.f*q.y);

  unsigned keys[TOPK];
  #pragma unroll
  for (int k = 0; k < TOPK; ++k) keys[k] = 0xFFFFFFFFu;

  int cbase = half * 8;            // per-lane column sub-offset inside a tile

  for (int ct = 0; ct < tiles; ++ct) {
    int cg = base + ct*16 + ln;    // A-matrix M=ln column node
    F3 cc = ldAtom(X, cg, 1);
    v2f af; // A 16x4: lanes 0-15 hold K=0,1; lanes 16-31 hold K=2,3 (ISA layout)
    af.x = half ? cc.z : cc.x;
    af.y = half ? 1.0f : cc.y;
    v8f acc;
    #pragma unroll
    for (int v = 0; v < 8; ++v)
      acc[v] = canorm[base + ct*16 + cbase + v]; // C[m][*] = |c_m|^2
    v8f d2 = __builtin_amdgcn_wmma_f32_16x16x4_f32(
        false, af, false, bf, (short)0, acc, false, false);
    #pragma unroll
    for (int v = 0; v < 8; ++v) {
      unsigned bits = __float_as_uint(fmaxf(d2[v], 0.f)); // >=0 => uint order == float order
      unsigned key  = (bits & ~IDXMASK) | (unsigned)(ct*16 + cbase + v);
      insertKey9(keys, key);
    }
  }

  // merge the two half-wave column ranges: partner = lane ^ 16 (index rides in the key)
  unsigned pk[TOPK];
  #pragma unroll
  for (int k = 0; k < TOPK; ++k)
    pk[k] = (unsigned)__shfl_xor((int)keys[k], 16, 32);
  #pragma unroll
  for (int k = 0; k < TOPK; ++k) insertKey9(keys, pk[k]);

  if (half == 0) {
    #pragma unroll
    for (int k = 0; k < TOPK; ++k) {
      size_t e = (size_t)qg * TOPK + k;
      erow[e]  = (float)qg;
      ecolf[e] = (float)(base + (int)(keys[k] & IDXMASK));
    }
  }
}

// ---------------- Kernel 3: per-edge features (bandwidth bound) ----------------
__global__ void k_edge(const float* __restrict__ X, const int* __restrict__ seg,
                       const float* __restrict__ Ofr, const float* __restrict__ ecolf,
                       float* __restrict__ ctx, float* __restrict__ inter,
                       float* __restrict__ edge_pos, float* __restrict__ dis_edge,
                       float* __restrict__ angle_edge, float* __restrict__ direct_edge,
                       int E)
{
  int e = blockIdx.x * blockDim.x + threadIdx.x;
  if (e >= E) return;
  int r = e / TOPK;
  int c = (int)ecolf[e];

  float same = (seg[r] == seg[c]) ? 1.f : 0.f;
  ctx[e]   = same;
  inter[e] = 1.f - same;

  { // edge positional embedding of (row - col)
    float t[16]; posembed16((float)(r - c), t);
    #pragma unroll
    for (int j = 0; j < 16; ++j) edge_pos[(size_t)e*16 + j] = t[j];
  }

  F3 xc1 = ldAtom(X, c, 1);

  // dis_edge: 4 atoms of row vs CA of col
  #pragma unroll
  for (int a = 0; a < 4; ++a) {
    F3 v = f3sub(ldAtom(X, r, a), xc1);
    float d = sqrtf(f3dot(v, v) + 1e-6f);
    float t[16]; rbf16(d, t);
    #pragma unroll
    for (int j = 0; j < 16; ++j) dis_edge[(size_t)e*64 + a*16 + j] = t[j];
  }

  float Or[9], Oc[9];
  #pragma unroll
  for (int k = 0; k < 9; ++k) { Or[k] = Ofr[(size_t)r*9 + k]; Oc[k] = Ofr[(size_t)c*9 + k]; }

  { // angle_edge: quaternion of R = Or^T * Oc
    float R[3][3];
    #pragma unroll
    for (int i2 = 0; i2 < 3; ++i2)
      #pragma unroll
      for (int j2 = 0; j2 < 3; ++j2)
        R[i2][j2] = Or[0*3+i2]*Oc[0*3+j2] + Or[1*3+i2]*Oc[1*3+j2] + Or[2*3+i2]*Oc[2*3+j2];
    float Rxx = R[0][0], Ryy = R[1][1], Rzz = R[2][2];
    float m0 = 0.5f * sqrtf(fabsf(1.f + Rxx - Ryy - Rzz) + 1e-12f);
    float m1 = 0.5f * sqrtf(fabsf(1.f - Rxx + Ryy - Rzz) + 1e-12f);
    float m2 = 0.5f * sqrtf(fabsf(1.f - Rxx - Ryy + Rzz) + 1e-12f);
    float qx = sgnf(R[2][1] - R[1][2]) * m0;
    float qy = sgnf(R[0][2] - R[2][0]) * m1;
    float qz = sgnf(R[1][0] - R[0][1]) * m2;
    float qw = sqrtf(fmaxf(1.f + Rxx + Ryy + Rzz, 0.f) + 1e-12f) * 0.5f;
    float n  = fmaxf(sqrtf(qx*qx + qy*qy + qz*qz + qw*qw), 1e-12f);
    angle_edge[(size_t)e*4 + 0] = qx / n;
    angle_edge[(size_t)e*4 + 1] = qy / n;
    angle_edge[(size_t)e*4 + 2] = qz / n;
    angle_edge[(size_t)e*4 + 3] = qw / n;
  }

  { // direct_edge: normalize(Oc @ (Xr[a] - Xc_CA)) for a in {0,2,3}
    const int atoms[3] = {0, 2, 3};
    #pragma unroll
    for (int a = 0; a < 3; ++a) {
      F3 v = f3sub(ldAtom(X, r, atoms[a]), xc1);
      F3 w = { Oc[0]*v.x + Oc[1]*v.y + Oc[2]*v.z,
               Oc[3]*v.x + Oc[4]*v.y + Oc[5]*v.z,
               Oc[6]*v.x + Oc[7]*v.y + Oc[8]*v.z };
      w = f3norm(w);
      direct_edge[(size_t)e*9 + a*3 + 0] = w.x;
      direct_edge[(size_t)e*9 + a*3 + 1] = w.y;
      direct_edge[(size_t)e*9 + a*3 + 2] = w.z;
    }
  }
}

extern "C" void kernel_launch(void* const* d_in, const int* in_sizes, int n_in,
                              void* d_out, int out_size, void* d_ws, size_t ws_size,
                              hipStream_t stream) {
  const float* X  = (const float*)d_in[0];
  const int* seg  = (const int*)d_in[1];
  int N = in_sizes[0] / 12;       // (N,4,3) float
  int L = N / 16;                 // B_BATCH = 16
  int E = N * TOPK;

  float* out = (float*)d_out;
  float* node_pos    = out;
  float* dis_node    = node_pos    + (size_t)N*16;
  float* angle_node  = dis_node    + (size_t)N*48;
  float* direct_node = angle_node  + (size_t)N*12;
  float* erow        = direct_node + (size_t)N*9;
  float* ecolf       = erow        + (size_t)E;
  float* ctx         = ecolf       + (size_t)E;
  float* inter       = ctx         + (size_t)E;
  float* edge_pos    = inter       + (size_t)E;
  float* dis_edge    = edge_pos    + (size_t)E*16;
  float* angle_edge  = dis_edge    + (size_t)E*64;
  float* direct_edge = angle_edge  + (size_t)E*4;

  float* canorm = (float*)d_ws;          // N floats
  float* Ofr    = canorm + N;            // N*9 floats

  int blk = 256;
  k_node<<<(N + blk - 1)/blk, blk, 0, stream>>>(
      X, node_pos, dis_node, angle_node, direct_node, canorm, Ofr, N, L);

  // one wave (32 lanes) per 16 query rows -> 2N threads total
  int topk_threads = 2 * N;
  k_topk<<<(topk_threads + blk - 1)/blk, blk, 0, stream>>>(
      X, canorm, erow, ecolf, N, L);

  k_edge<<<(E + blk - 1)/blk, blk, 0, stream>>>(
      X, seg, Ofr, ecolf, ctx, inter, edge_pos, dis_edge, angle_edge, direct_edge, E);
}